// Conduits_34127810134288
// MI455X (gfx1250) — compile-verified
//
#include <hip/hip_runtime.h>
#include <hip/hip_bf16.h>
#include <math.h>

typedef float f32x4 __attribute__((ext_vector_type(4)));
typedef int   i32x4 __attribute__((ext_vector_type(4)));

#define G_ACC    9.81f
#define RHO_I    917.0f
#define RHO_W    1000.0f
#define NU_W     1.787e-6f
#define OMEGA_F  1e-3f
#define LHEAT_F  334000.0f
#define AFLU_F   6e-24f
#define U0_F     50.0f
#define TAN_PHI  0.6248693519093275f   // tan(deg2rad(32))

// Fast base-2 transcendentals: v_exp_f32 / v_log_f32 on gfx1250.
__device__ __forceinline__ float fast_exp2(float x) {
#if __has_builtin(__builtin_amdgcn_exp2f)
    return __builtin_amdgcn_exp2f(x);
#else
    return exp2f(x);
#endif
}
__device__ __forceinline__ float fast_log2(float x) {
#if __has_builtin(__builtin_amdgcn_logf)
    return __builtin_amdgcn_logf(x);   // hardware log2
#else
    return log2f(x);
#endif
}

// ---------------------------------------------------------------------------
// Kernel 1: per-link. grad_h = (h[head]-h[tail])/len ; Q = -T*grad_h.
// Pack {grad_h, Q, u_link, 0} into one float4 so the node kernel needs only
// ONE b128 gather per link slot. Streaming inputs are non-temporal so the
// 16 MB h[] gather target stays resident in the 192 MB L2.
// ---------------------------------------------------------------------------
__global__ __launch_bounds__(256) void link_kernel(
    const float* __restrict__ conduit, const float* __restrict__ rey,
    const float* __restrict__ usl,     const float* __restrict__ len,
    const float* __restrict__ h,
    const int*   __restrict__ head,    const int* __restrict__ tail,
    f32x4* __restrict__ ws, int L)
{
    int i = blockIdx.x * blockDim.x + threadIdx.x;
    if (i >= L) return;

    int   hn = __builtin_nontemporal_load(head + i);
    int   tn = __builtin_nontemporal_load(tail + i);
    float ll = __builtin_nontemporal_load(len  + i);

    float gh = (h[hn] - h[tn]) / ll;               // L2-resident gathers

    float cs = __builtin_nontemporal_load(conduit + i);
    float re = __builtin_nontemporal_load(rey     + i);
    float T  = (cs * cs * cs) * (G_ACC / (12.0f * NU_W)) / (1.0f + OMEGA_F * re);
    float Q  = -T * gh;
    float u  = __builtin_nontemporal_load(usl + i);

    f32x4 v; v.x = gh; v.y = Q; v.z = u; v.w = 0.0f;
    ws[i] = v;                                     // regular store: keep dirty in L2
}

// ---------------------------------------------------------------------------
// Common per-node tail math (shared by fast and fallback paths).
// ---------------------------------------------------------------------------
__device__ __forceinline__ float node_tail(
    float grad_n, float Q_n, float u_n, float flux,
    float hh, float thick, float bed, float melt_in, float geo, float area)
{
    float ob   = RHO_I * G_ACC * thick;
    float pw   = fminf((hh - bed) * (RHO_W * G_ACC), ob);
    float Neff = ob - pw;

    float ua    = fabsf(u_n);
    float ratio = ua / (ua + U0_F);
    float tau   = Neff * TAN_PHI * fast_exp2(0.2f * fast_log2(ratio));

    float fric = fabsf(u_n * tau);
    float diss = RHO_W * G_ACC * Q_n * grad_n;
    float melt = (geo + fric - diss) * (1.0f / LHEAT_F);

    float flux_term = flux / area;
    float melt_term = -melt * (1.0f / RHO_W - 1.0f / RHO_I);
    float closure   = -AFLU_F * (Neff * Neff * Neff) * hh;
    return flux_term - melt_term - closure - melt_in + hh;
}

// ---------------------------------------------------------------------------
// Kernel 2: per-node. 4 gathers of b128 from the L2-resident workspace;
// everything else is coalesced non-temporal streaming.
// ---------------------------------------------------------------------------
__global__ __launch_bounds__(256) void node_kernel(
    const f32x4* __restrict__ ws,
    const float* __restrict__ h,       const float* __restrict__ thick,
    const float* __restrict__ bed,     const float* __restrict__ melt_in,
    const float* __restrict__ geo,     const float* __restrict__ area,
    const f32x4* __restrict__ dirs,    const i32x4* __restrict__ links,
    float* __restrict__ out, int N)
{
    int n = blockIdx.x * blockDim.x + threadIdx.x;
    if (n >= N) return;

    i32x4 lk = __builtin_nontemporal_load(links + n);
    f32x4 dr = __builtin_nontemporal_load(dirs  + n);

    f32x4 a = ws[lk.x];
    f32x4 b = ws[lk.y];
    f32x4 c = ws[lk.z];
    f32x4 d = ws[lk.w];

    float grad_n = 0.25f * (a.x + b.x + c.x + d.x);
    float Q_n    = 0.25f * (a.y + b.y + c.y + d.y);
    float u_n    = 0.25f * (a.z + b.z + c.z + d.z);
    float flux   = dr.x * a.y + dr.y * b.y + dr.z * c.y + dr.w * d.y;

    float hh = __builtin_nontemporal_load(h       + n);
    float th = __builtin_nontemporal_load(thick   + n);
    float be = __builtin_nontemporal_load(bed     + n);
    float mi = __builtin_nontemporal_load(melt_in + n);
    float ge = __builtin_nontemporal_load(geo     + n);
    float ar = __builtin_nontemporal_load(area    + n);

    float res = node_tail(grad_n, Q_n, u_n, flux, hh, th, be, mi, ge, ar);
    __builtin_nontemporal_store(res, out + n);
}

// ---------------------------------------------------------------------------
// Fallback (only if the harness workspace is smaller than 8M * 16B):
// fully recompute link quantities per node. Correct, slower.
// ---------------------------------------------------------------------------
__global__ __launch_bounds__(256) void node_fallback_kernel(
    const float* __restrict__ conduit, const float* __restrict__ rey,
    const float* __restrict__ usl,     const float* __restrict__ len,
    const float* __restrict__ h,       const float* __restrict__ thick,
    const float* __restrict__ bed,     const float* __restrict__ melt_in,
    const float* __restrict__ geo,     const float* __restrict__ area,
    const f32x4* __restrict__ dirs,
    const int* __restrict__ head,      const int* __restrict__ tail,
    const i32x4* __restrict__ links,
    float* __restrict__ out, int N)
{
    int n = blockIdx.x * blockDim.x + threadIdx.x;
    if (n >= N) return;

    i32x4 lk = links[n];
    f32x4 dr = dirs[n];
    int   li[4] = { lk.x, lk.y, lk.z, lk.w };
    float dv[4] = { dr.x, dr.y, dr.z, dr.w };

    float gsum = 0.f, qsum = 0.f, usum = 0.f, flux = 0.f;
#pragma unroll
    for (int k = 0; k < 4; ++k) {
        int   l  = li[k];
        float gh = (h[head[l]] - h[tail[l]]) / len[l];
        float cs = conduit[l];
        float T  = (cs * cs * cs) * (G_ACC / (12.0f * NU_W)) / (1.0f + OMEGA_F * rey[l]);
        float Q  = -T * gh;
        gsum += gh; qsum += Q; usum += usl[l]; flux += dv[k] * Q;
    }
    float res = node_tail(0.25f * gsum, 0.25f * qsum, 0.25f * usum, flux,
                          h[n], thick[n], bed[n], melt_in[n], geo[n], area[n]);
    out[n] = res;
}

// ---------------------------------------------------------------------------
// Compile probe (never launched): verify the gfx1250 async-to-LDS assembler
// path on this toolchain. Candidate for software-pipelined staging of the
// streaming link arrays in link_kernel if it assembles cleanly.
// ---------------------------------------------------------------------------
__global__ void __launch_bounds__(32) cdna5_async_probe(const float* __restrict__ g,
                                                        float* __restrict__ o)
{
    __shared__ float lds[32];
    unsigned lds_byte    = threadIdx.x * 4u;
    const float* gptr    = g + threadIdx.x;
    asm volatile("global_load_async_to_lds_b32 %0, %1, off"
                 :: "v"(lds_byte), "v"(gptr) : "memory");
    asm volatile("s_wait_asynccnt 0" ::: "memory");
    __syncthreads();
    o[threadIdx.x] = lds[threadIdx.x];
}

// ---------------------------------------------------------------------------
extern "C" void kernel_launch(void* const* d_in, const int* in_sizes, int n_in,
                              void* d_out, int out_size, void* d_ws, size_t ws_size,
                              hipStream_t stream)
{
    const float* conduit = (const float*)d_in[0];
    const float* rey     = (const float*)d_in[1];
    const float* usl     = (const float*)d_in[2];
    const float* len     = (const float*)d_in[3];
    const float* h       = (const float*)d_in[4];
    const float* thick   = (const float*)d_in[5];
    const float* bed     = (const float*)d_in[6];
    const float* melt_in = (const float*)d_in[7];
    const float* geo     = (const float*)d_in[8];
    const float* area    = (const float*)d_in[9];
    const f32x4* dirs    = (const f32x4*)d_in[10];   // [N,4]
    const int*   head    = (const int*)d_in[11];
    const int*   tail    = (const int*)d_in[12];
    const i32x4* links   = (const i32x4*)d_in[13];   // [N,4]
    float*       out     = (float*)d_out;

    const int L = in_sizes[0];   // 8M links
    const int N = in_sizes[4];   // 4M nodes

    const size_t need = (size_t)L * sizeof(f32x4);   // 128 MB packed link records

    if (ws_size >= need) {
        f32x4* ws = (f32x4*)d_ws;
        link_kernel<<<(L + 255) / 256, 256, 0, stream>>>(
            conduit, rey, usl, len, h, head, tail, ws, L);
        node_kernel<<<(N + 255) / 256, 256, 0, stream>>>(
            ws, h, thick, bed, melt_in, geo, area, dirs, links, out, N);
    } else {
        node_fallback_kernel<<<(N + 255) / 256, 256, 0, stream>>>(
            conduit, rey, usl, len, h, thick, bed, melt_in, geo, area,
            dirs, head, tail, links, out, N);
    }
}